// SphericalExpansionCalculator_54700703482415
// MI455X (gfx1250) — compile-verified
//
#include <hip/hip_runtime.h>
#include <hip/hip_bf16.h>

typedef float v2f __attribute__((ext_vector_type(2)));
typedef float v8f __attribute__((ext_vector_type(8)));

#define N_COMP   16   // sum of (2l+1) for l=0..3
#define N_RADIAL 20
#define N_PSEUDO 4
#define N_SPEC   4
#define ELEMS    (N_COMP * N_RADIAL)            // 320 per pair
#define FEAT     (N_COMP * N_PSEUDO * N_RADIAL) // 1280 per center
#define TILES_PER_CENTER (ELEMS / 16)           // 20 WMMA row-tiles per center

// 32-bit byte-offset addressing: base is uniform (SGPR pair), offset is a
// 32-bit VGPR -> backend emits GLOBAL_*_SADDR (GVS mode), no 64-bit VALU adds.
__device__ __forceinline__ const float* fptr(const float* base, unsigned eidx) {
    return (const float*)((const char*)base + (eidx << 2));
}
__device__ __forceinline__ float* fptr(float* base, unsigned eidx) {
    return (float*)((char*)base + (eidx << 2));
}

// ---------------------------------------------------------------------------
// Phase 1: scatter-add vec values into partial[center][c_glob*20+n][species].
// One block per pair (320 threads = one (c_glob, n) element each).
// Multiplication by W is deferred to phase 2 (4x fewer atomics), and the
// atomic target (51 MB) stays resident in the 192 MB L2.
// ---------------------------------------------------------------------------
__global__ __launch_bounds__(ELEMS) void se_scatter_kernel(
    const float* __restrict__ v0, const float* __restrict__ v1,
    const float* __restrict__ v2, const float* __restrict__ v3,
    const int*   __restrict__ species, const int* __restrict__ centers,
    float* __restrict__ partial)
{
    const unsigned p  = blockIdx.x;
    const unsigned j  = threadIdx.x;   // 0..319 == c_glob*20 + n
    const unsigned cg = j / N_RADIAL;
    const unsigned n  = j % N_RADIAL;

    const float* vp; unsigned cl, ncomp;
    if (cg < 1)      { vp = v0; cl = cg;     ncomp = 1; }
    else if (cg < 4) { vp = v1; cl = cg - 1; ncomp = 3; }
    else if (cg < 9) { vp = v2; cl = cg - 4; ncomp = 5; }
    else             { vp = v3; cl = cg - 9; ncomp = 7; }

    // streaming read: touched exactly once, keep it out of L2's way
    const unsigned vidx = (p * ncomp + cl) * N_RADIAL + n;    // < 2^31 bytes
    const float v = __builtin_nontemporal_load(fptr(vp, vidx));
    const int   s = species[p];   // uniform per block -> scalar load
    const int   c = centers[p];

    // partial row = c*320 + j, innermost dim = species (K of the phase-2 GEMM)
    const unsigned pidx = ((unsigned)c * ELEMS + j) * N_SPEC + (unsigned)s;
    atomicAdd(fptr(partial, pidx), v);
}

// ---------------------------------------------------------------------------
// Phase 2: dens[row, d] = sum_s partial[row, s] * W[d, s]
// One wave handles 16 rows via V_WMMA_F32_16X16X4_F32:
//   A (16x4)  = 16 partial rows, K = species
//   B (4x16)  = W^T in columns 0..3, zeros elsewhere
//   D (16x16) = columns 0..3 hold the 4 pseudo channels
// 320 rows/center = 20 tiles/center, so a tile never straddles a center:
// center / rem0 are wave-uniform (pure SALU).
// ---------------------------------------------------------------------------
__global__ __launch_bounds__(256) void se_gemm_kernel(
    const float* __restrict__ partial, const float* __restrict__ W,
    float* __restrict__ out, int nwaves)
{
    const int wave = (int)((blockIdx.x * blockDim.x + threadIdx.x) >> 5);
    if (wave >= nwaves) return;              // wave-uniform: EXEC stays all-1s

    const unsigned lane = threadIdx.x & 31;
    const unsigned m    = lane & 15;         // row in tile (A) / col (B,D)
    const unsigned k0   = (lane >> 4) << 1;  // K pair: {0,1} or {2,3}

    // wave-uniform tile -> center decomposition (SALU)
    const unsigned uw     = (unsigned)wave;
    const unsigned center = uw / TILES_PER_CENTER;
    const unsigned rem0   = (uw % TILES_PER_CENTER) * 16;  // row offset in center

    // A operand: 16 rows x 4 species, one aligned b64 saddr load per lane
    const unsigned aidx = (center * ELEMS + rem0 + m) * N_SPEC + k0;
    v2f a = *(const v2f*)fptr(partial, aidx);

    // B operand: unconditional clamped b64 load, then value-select (no EXEC branch)
    const unsigned bm = (m < N_PSEUDO) ? m : 0u;
    v2f bw = *(const v2f*)fptr(W, bm * N_SPEC + k0);
    v2f b;
    b.x = (m < N_PSEUDO) ? bw.x : 0.0f;
    b.y = (m < N_PSEUDO) ? bw.y : 0.0f;

    v8f c = {};
    c = __builtin_amdgcn_wmma_f32_16x16x4_f32(
        /*neg_a=*/false, a, /*neg_b=*/false, b,
        /*c_mod=*/(short)0, c, /*reuse_a=*/false, /*reuse_b=*/false);

    // D layout: VGPR r, lane: row = r + (lane>=16 ? 8 : 0), col = lane&15.
    // Only cols 0..3 (pseudo channels) are stored; out is write-once -> NT store.
    if (m < N_PSEUDO) {
        const unsigned d     = m;
        const unsigned half8 = (lane >= 16) ? 8u : 0u;
        const unsigned obase = center * FEAT + d * N_RADIAL;
        const unsigned rbase = rem0 + half8;
        #pragma unroll
        for (unsigned r = 0; r < 8; ++r) {
            const unsigned rem = rbase + r;          // cg*20 + n, 0..319
            const unsigned cg  = rem / N_RADIAL;     // small magic-mul div
            const unsigned n   = rem - cg * N_RADIAL;
            __builtin_nontemporal_store(c[r],
                fptr(out, obase + cg * (N_PSEUDO * N_RADIAL) + n));
        }
    }
}

extern "C" void kernel_launch(void* const* d_in, const int* in_sizes, int n_in,
                              void* d_out, int out_size, void* d_ws, size_t ws_size,
                              hipStream_t stream) {
    const float* v0      = (const float*)d_in[0];
    const float* v1      = (const float*)d_in[1];
    const float* v2      = (const float*)d_in[2];
    const float* v3      = (const float*)d_in[3];
    const float* W       = (const float*)d_in[4];
    const int*   species = (const int*)d_in[5];
    const int*   centers = (const int*)d_in[6];
    float*       out     = (float*)d_out;

    const int n_pairs   = in_sizes[0] / N_RADIAL;   // vec_l0 is [P, 1, 20]
    const int n_centers = out_size / FEAT;

    float* partial = (float*)d_ws;
    const size_t partial_bytes = (size_t)n_centers * FEAT * sizeof(float);

    hipMemsetAsync(partial, 0, partial_bytes, stream);

    se_scatter_kernel<<<n_pairs, ELEMS, 0, stream>>>(
        v0, v1, v2, v3, species, centers, partial);

    const int nwaves  = n_centers * TILES_PER_CENTER;  // 16 rows per wave
    const int threads = 256;                           // 8 waves per block
    const int blocks  = (nwaves * 32 + threads - 1) / threads;
    se_gemm_kernel<<<blocks, threads, 0, stream>>>(partial, W, out, nwaves);
}